// DenseEquivariantFFT_35253091565743
// MI455X (gfx1250) — compile-verified
//
#include <hip/hip_runtime.h>
#include <hip/hip_bf16.h>

// DenseEquivariantFFT on gfx1250 (MI455X), wave32 + V_WMMA_F32_16X16X4_F32.
//
// Pipeline (all fp32, FFT via 16x16 DFT-matrix WMMA products):
//   1) fft_x : x(B=64,Cin=64,2048) -> Xf[k][b][ip]       (complex, k=256 freqs)
//   2) fft_k : kernel gathered via product_table -> Wf[k][ip][fq]
//   3) gemm  : per-frequency complex GEMM Y[k] = X[k] (64x512) * W[k] (512x512)
//   4) ifft_y: inverse 2D FFT (real part) * 1/256 + bias -> out(B,F,2048)
//
// Workspace (floats): Xr,Xi (8.39M each) Wr,Wi (67.1M each) Yr,Yi (8.39M each)
// = 167.8M floats = 671 MB of d_ws.

typedef __attribute__((ext_vector_type(2))) float v2f;
typedef __attribute__((ext_vector_type(8))) float v8f;

__device__ __forceinline__ v8f wmma4(v2f a, v2f b, v8f c) {
  // D = A(16x4) * B(4x16) + C(16x16), fp32
  return __builtin_amdgcn_wmma_f32_16x16x4_f32(false, a, false, b, (short)0, c,
                                               false, false);
}

#define TWO_PI_OVER_16 0.39269908169872414f

// Per-lane fragment of the symmetric 16x16 DFT matrices C=cos, S=sin for the
// WMMA K-slice t (K = 4t..4t+3). Because C,S are symmetric, the same lane
// values serve as both A-fragments and B-fragments.
__device__ __forceinline__ void dft_frag(int t, int m, int hi, v2f& c, v2f& s) {
  int k0 = 4 * t + 2 * hi;
  float a0 = (float)((m * k0) & 15) * TWO_PI_OVER_16;
  float a1 = (float)((m * (k0 + 1)) & 15) * TWO_PI_OVER_16;
  c.x = __cosf(a0); c.y = __cosf(a1);
  s.x = __sinf(a0); s.y = __sinf(a1);
}

// ---------------------------------------------------------------------------
// Forward 2D FFT of one 16x16 real tile per wave:  X = F D F,  F = C - iS
// Stage 1: Tr = C*D, Ti = -S*D   Stage 2: Xr = Tr*C + Ti*S, Xi = Ti*C - Tr*S
// ---------------------------------------------------------------------------
__device__ __forceinline__ void fft_tile(float* l0, float* l1, int lane,
                                         v8f& xr_out, v8f& xi_out) {
  const int m = lane & 15;
  const int hi = lane >> 4;
  v2f Db[4], Ca[4], Sa[4], Sn[4];
#pragma unroll
  for (int t = 0; t < 4; ++t) {
    int k0 = 4 * t + 2 * hi;
    Db[t].x = l0[k0 * 16 + m];
    Db[t].y = l0[(k0 + 1) * 16 + m];
    dft_frag(t, m, hi, Ca[t], Sa[t]);
    Sn[t] = -Sa[t];
  }
  v8f Tr = {}; v8f Ti = {};
#pragma unroll
  for (int t = 0; t < 4; ++t) Tr = wmma4(Ca[t], Db[t], Tr);
#pragma unroll
  for (int t = 0; t < 4; ++t) Ti = wmma4(Sn[t], Db[t], Ti);
  __syncthreads();
#pragma unroll
  for (int r = 0; r < 8; ++r) {
    l0[(r + hi * 8) * 16 + m] = Tr[r];
    l1[(r + hi * 8) * 16 + m] = Ti[r];
  }
  __syncthreads();
  v2f TrA[4], TiA[4], TrN[4];
#pragma unroll
  for (int t = 0; t < 4; ++t) {
    int k0 = 4 * t + 2 * hi;
    TrA[t].x = l0[m * 16 + k0]; TrA[t].y = l0[m * 16 + k0 + 1];
    TiA[t].x = l1[m * 16 + k0]; TiA[t].y = l1[m * 16 + k0 + 1];
    TrN[t] = -TrA[t];
  }
  v8f xr = {}; v8f xi = {};
#pragma unroll
  for (int t = 0; t < 4; ++t) {
    xr = wmma4(TrA[t], Ca[t], xr);
    xr = wmma4(TiA[t], Sa[t], xr);
    xi = wmma4(TiA[t], Ca[t], xi);
    xi = wmma4(TrN[t], Sa[t], xi);
  }
  xr_out = xr; xi_out = xi;
}

// ------------------------- kernel 1: FFT of x ------------------------------
__global__ __launch_bounds__(256) void fft_x_kernel(
    const float* __restrict__ x, float* __restrict__ Xr, float* __restrict__ Xi) {
  __shared__ float lds[2][8][256];
  const int lane = threadIdx.x & 31;
  const int wave = threadIdx.x >> 5;
  const int tile = blockIdx.x * 8 + wave;   // 0..32767  (b, i, p)
  const int b = tile >> 9;
  const int ip = tile & 511;
  const int i = ip >> 3;
  const int p = ip & 7;
  float* l0 = lds[0][wave];
  float* l1 = lds[1][wave];
  const float* src = x + (size_t)(b * 64 + i) * 2048 + p;
#pragma unroll
  for (int r = 0; r < 8; ++r) l0[r * 32 + lane] = src[(size_t)(r * 32 + lane) * 8];
  __syncthreads();
  v8f xr, xi;
  fft_tile(l0, l1, lane, xr, xi);
  const int m = lane & 15;
  const int hi = lane >> 4;
  const size_t obase = (size_t)b * 512 + ip;   // within a [k] plane (64*512)
#pragma unroll
  for (int r = 0; r < 8; ++r) {
    int k = ((r + hi * 8) * 16 + m);           // freq cell ju*16+jv
    Xr[(size_t)k * 32768 + obase] = xr[r];
    Xi[(size_t)k * 32768 + obase] = xi[r];
  }
}

// ---------------- kernel 2: gather + FFT of weight kernel ------------------
__global__ __launch_bounds__(256) void fft_k_kernel(
    const float* __restrict__ kern, const int* __restrict__ ptab,
    float* __restrict__ Wr, float* __restrict__ Wi) {
  __shared__ float lds[2][8][256];
  const int lane = threadIdx.x & 31;
  const int wave = threadIdx.x >> 5;
  const int tile = blockIdx.x * 8 + wave;   // 0..262143  (f, i, p, q)
  const int q = tile & 7;
  const int p = (tile >> 3) & 7;
  const int i = (tile >> 6) & 63;
  const int f = tile >> 12;
  float* l0 = lds[0][wave];
  float* l1 = lds[1][wave];
  const float* src = kern + (size_t)(f * 64 + i) * 2048;
  const int* pt = ptab + p * 2048 + q;       // pt[cell*8] = mapping index
#pragma unroll
  for (int r = 0; r < 8; ++r) {
    int cell = r * 32 + lane;
    l0[cell] = src[pt[cell * 8]];
  }
  __syncthreads();
  v8f wr, wi;
  fft_tile(l0, l1, lane, wr, wi);
  const int m = lane & 15;
  const int hi = lane >> 4;
  const size_t obase = (size_t)(i * 8 + p) * 512 + (f * 8 + q); // within [k] plane
#pragma unroll
  for (int r = 0; r < 8; ++r) {
    int k = ((r + hi * 8) * 16 + m);
    Wr[(size_t)k * 262144 + obase] = wr[r];
    Wi[(size_t)k * 262144 + obase] = wi[r];
  }
}

// ------------- kernel 3: per-frequency complex GEMM (WMMA f32) -------------
// Y[k] (64x512) = X[k] (64x512) * W[k] (512x512), complex.
// 128 consecutive waves share one k so its 2 MB W-slice stays L2-resident.
__global__ __launch_bounds__(256) void gemm_kernel(
    const float* __restrict__ Xr, const float* __restrict__ Xi,
    const float* __restrict__ Wr, const float* __restrict__ Wi,
    float* __restrict__ Yr, float* __restrict__ Yi) {
  const int lane = threadIdx.x & 31;
  const int wave = threadIdx.x >> 5;
  const int w = blockIdx.x * 8 + wave;      // 0..32767
  const int k = w >> 7;                     // 256 freqs
  const int rem = w & 127;
  const int mt = rem >> 5;                  // 4 M-tiles (batch)
  const int nt = rem & 31;                  // 32 N-tiles (f*8+q)
  const int m = lane & 15;
  const int hi = lane >> 4;
  const float* xr = Xr + (size_t)k * 32768 + (size_t)(mt * 16 + m) * 512;
  const float* xi = Xi + (size_t)k * 32768 + (size_t)(mt * 16 + m) * 512;
  const float* wr = Wr + (size_t)k * 262144 + nt * 16 + m;
  const float* wi = Wi + (size_t)k * 262144 + nt * 16 + m;
  v8f aR = {}; v8f aI = {};
  for (int kk = 0; kk < 512; kk += 4) {
    const int k0 = kk + 2 * hi;
    v2f ar = *(const v2f*)(xr + k0);
    v2f ai = *(const v2f*)(xi + k0);
    v2f br, bi;
    br.x = wr[(size_t)k0 * 512];  br.y = wr[(size_t)(k0 + 1) * 512];
    bi.x = wi[(size_t)k0 * 512];  bi.y = wi[(size_t)(k0 + 1) * 512];
    v2f an = -ai;
    aR = wmma4(ar, br, aR);   // Re += Xr*Wr
    aR = wmma4(an, bi, aR);   // Re -= Xi*Wi
    aI = wmma4(ar, bi, aI);   // Im += Xr*Wi
    aI = wmma4(ai, br, aI);   // Im += Xi*Wr
  }
  float* yr = Yr + (size_t)k * 32768;
  float* yi = Yi + (size_t)k * 32768;
#pragma unroll
  for (int r = 0; r < 8; ++r) {
    size_t off = (size_t)(mt * 16 + r + hi * 8) * 512 + nt * 16 + m;
    yr[off] = aR[r];
    yi[off] = aI[r];
  }
}

// ---------------- kernel 4: inverse FFT (real part) + bias -----------------
// y = (1/256) * Re[ conj(F) * Yf * conj(F) ],  conj(F) = C + iS
__global__ __launch_bounds__(256) void ifft_y_kernel(
    const float* __restrict__ Yr, const float* __restrict__ Yi,
    const float* __restrict__ bias, float* __restrict__ out) {
  __shared__ float lds[2][8][256];
  const int lane = threadIdx.x & 31;
  const int wave = threadIdx.x >> 5;
  const int tile = blockIdx.x * 8 + wave;   // 0..32767  (b, f, q)
  const int q = tile & 7;
  const int f = (tile >> 3) & 63;
  const int b = tile >> 9;
  float* l0 = lds[0][wave];
  float* l1 = lds[1][wave];
  const size_t ibase = (size_t)b * 512 + f * 8 + q;  // within a [k] plane
#pragma unroll
  for (int r = 0; r < 8; ++r) {
    int cell = r * 32 + lane;                        // freq ju*16+jv
    l0[cell] = Yr[(size_t)cell * 32768 + ibase];
    l1[cell] = Yi[(size_t)cell * 32768 + ibase];
  }
  __syncthreads();
  const int m = lane & 15;
  const int hi = lane >> 4;
  v2f YrB[4], YiB[4], Ca[4], Sa[4], Sn[4];
#pragma unroll
  for (int t = 0; t < 4; ++t) {
    int k0 = 4 * t + 2 * hi;
    YrB[t].x = l0[k0 * 16 + m];  YrB[t].y = l0[(k0 + 1) * 16 + m];
    YiB[t].x = l1[k0 * 16 + m];  YiB[t].y = l1[(k0 + 1) * 16 + m];
    dft_frag(t, m, hi, Ca[t], Sa[t]);
    Sn[t] = -Sa[t];
  }
  v8f Tr = {}; v8f Ti = {};
#pragma unroll
  for (int t = 0; t < 4; ++t) {
    Tr = wmma4(Ca[t], YrB[t], Tr);   // C*Yr
    Tr = wmma4(Sn[t], YiB[t], Tr);   // -S*Yi
    Ti = wmma4(Ca[t], YiB[t], Ti);   // C*Yi
    Ti = wmma4(Sa[t], YrB[t], Ti);   // +S*Yr
  }
  __syncthreads();
#pragma unroll
  for (int r = 0; r < 8; ++r) {
    l0[(r + hi * 8) * 16 + m] = Tr[r];
    l1[(r + hi * 8) * 16 + m] = Ti[r];
  }
  __syncthreads();
  v2f TrA[4], TiN[4];
#pragma unroll
  for (int t = 0; t < 4; ++t) {
    int k0 = 4 * t + 2 * hi;
    TrA[t].x = l0[m * 16 + k0];  TrA[t].y = l0[m * 16 + k0 + 1];
    v2f ti;
    ti.x = l1[m * 16 + k0];      ti.y = l1[m * 16 + k0 + 1];
    TiN[t] = -ti;
  }
  v8f y = {};
#pragma unroll
  for (int t = 0; t < 4; ++t) {
    y = wmma4(TrA[t], Ca[t], y);     // Tr*C
    y = wmma4(TiN[t], Sa[t], y);     // -Ti*S
  }
  const float bf = bias[f];
  float* dst = out + (size_t)(b * 64 + f) * 2048 + q;
#pragma unroll
  for (int r = 0; r < 8; ++r) {
    int cell = (r + hi * 8) * 16 + m;            // spatial tu*16+tv
    dst[(size_t)cell * 8] = y[r] * (1.0f / 256.0f) + bf;
  }
}

extern "C" void kernel_launch(void* const* d_in, const int* in_sizes, int n_in,
                              void* d_out, int out_size, void* d_ws, size_t ws_size,
                              hipStream_t stream) {
  const float* x    = (const float*)d_in[0];   // (64, 64, 2048)
  const float* kern = (const float*)d_in[1];   // (64, 64, 2048)
  const float* bias = (const float*)d_in[2];   // (64,)
  const int*   ptab = (const int*)d_in[3];     // (8, 2048) int

  float* out = (float*)d_out;                  // (64, 64, 2048)
  float* ws  = (float*)d_ws;

  const size_t SX = (size_t)256 * 64 * 512;    //  8.39M floats
  const size_t SW = (size_t)256 * 512 * 512;   // 67.1M floats
  float* Xr = ws;
  float* Xi = Xr + SX;
  float* Wr = Xi + SX;
  float* Wi = Wr + SW;
  float* Yr = Wi + SW;
  float* Yi = Yr + SX;

  fft_x_kernel<<<4096, 256, 0, stream>>>(x, Xr, Xi);
  fft_k_kernel<<<32768, 256, 0, stream>>>(kern, ptab, Wr, Wi);
  gemm_kernel<<<4096, 256, 0, stream>>>(Xr, Xi, Wr, Wi, Yr, Yi);
  ifft_y_kernel<<<4096, 256, 0, stream>>>(Yr, Yi, bias, out);
}